// NotGPT_10557029613765
// MI455X (gfx1250) — compile-verified
//
#include <hip/hip_runtime.h>
#include <math.h>

// Problem dims (compile-time constants from the reference)
#define BB   2048
#define TT   50
#define VOC  65
#define EMB  100      // = H*D
#define HEADS 20
#define HS   5
#define BT   (BB*TT)  // 102400 rows
#define MBLK 128      // rows per workgroup (8 waves x 16)
#define NBLOCKS (BT/MBLK)  // 800
#define LDA  104      // LDS stride for A stage (pad vs bank conflicts)

#define ASG __attribute__((address_space(1)))   // global  (prints as __device__)
#define ASL __attribute__((address_space(3)))   // LDS     (prints as __shared__)

typedef int v4i __attribute__((__vector_size__(4 * sizeof(int))));

#if __has_builtin(__builtin_amdgcn_global_load_async_to_lds_b128) && \
    __has_builtin(__builtin_amdgcn_global_load_async_to_lds_b32) &&  \
    __has_builtin(__builtin_amdgcn_s_wait_asynccnt)
#define USE_ASYNC_LDS 1
#else
#define USE_ASYNC_LDS 0
#endif

typedef __attribute__((ext_vector_type(2))) float v2f;
typedef __attribute__((ext_vector_type(8))) float v8f;

__device__ __forceinline__ v8f wmma4(v2f a, v2f b, v8f c) {
  // D = A(16x4 f32) * B(4x16 f32) + C(16x16 f32)
  return __builtin_amdgcn_wmma_f32_16x16x4_f32(
      /*neg_a=*/false, a, /*neg_b=*/false, b,
      /*c_mod=*/(short)0, c, /*reuse_a=*/false, /*reuse_b=*/false);
}

// Stage 128 rows x 100 cols of A into LDS (stride LDA). Both the global rows
// (400 B stride) and LDS rows (416 B stride) are 16B aligned, so we move
// 16 B per op. Preferred path: async global->LDS DMA (no VGPR round trip),
// tracked by ASYNCcnt; fallback: vectorized float4 copy.
__device__ __forceinline__ void stage_A(const float* __restrict__ A,
                                        float* As, int mbase, int tid) {
  for (int c = tid; c < MBLK * 25; c += 256) {
    const int r = c / 25;
    const int q = c - r * 25;           // 16-byte chunk within the row
    const float* g = A + (size_t)(mbase + r) * EMB + q * 4;
    float* l = As + r * LDA + q * 4;
#if USE_ASYNC_LDS
    __builtin_amdgcn_global_load_async_to_lds_b128(
        (ASG v4i*)g, (ASL v4i*)l, /*offset=*/0, /*cpol=*/0);
#else
    *(float4*)l = *(const float4*)g;
#endif
  }
#if USE_ASYNC_LDS
  __builtin_amdgcn_s_wait_asynccnt(0);
#endif
  __syncthreads();
}

// ---------------------------------------------------------------------------
// x[bt,e] = tok_emb[idx[bt], e] + pos_emb[bt%T, e]
// ---------------------------------------------------------------------------
__global__ __launch_bounds__(256) void k_embed(const int* __restrict__ idx,
                                               const float* __restrict__ tok,
                                               const float* __restrict__ pos,
                                               float* __restrict__ x) {
  int i = blockIdx.x * 256 + threadIdx.x;
  if (i >= BT * EMB) return;
  int bt = i / EMB;
  int e  = i - bt * EMB;
  int t  = bt % TT;
  x[i] = tok[idx[bt] * EMB + e] + pos[t * EMB + e];
}

// ---------------------------------------------------------------------------
// Zero-pad W [N x EMB] to Wp [rows x EMB] (rows = NT*16) so the GEMM inner
// loop needs no per-lane bounds check (keeps EXEC full, no exec save/restore).
// ---------------------------------------------------------------------------
__global__ __launch_bounds__(256) void k_padw(const float* __restrict__ W,
                                              float* __restrict__ Wp,
                                              int N, int rows) {
  int i = blockIdx.x * 256 + threadIdx.x;
  if (i >= rows * EMB) return;
  int r = i / EMB;
  int c = i - r * EMB;
  Wp[i] = (r < N) ? W[r * EMB + c] : 0.f;
}

// ---------------------------------------------------------------------------
// C[m,n] = act( sum_k A[m,k] * Wp[n,k] + bias[n] ),  K = EMB = 100, ldc = N.
// One workgroup = 8 waves = 128 rows of A staged in LDS. Each wave computes
// one 16-row M-tile across NT 16-col N-tiles, keeping NT v8f accumulators in
// VGPRs so the A fragment is fetched once per K step (ds_load_b64) and reused.
// Wp is pre-padded: all B loads are unconditional global_load_b64.
// Fragment layouts per CDNA5 ISA 7.12.2:
//   A 16x4 : lanes 0-15 -> {K0,K1}, lanes 16-31 -> {K2,K3}, M = lane%16
//   B 4x16 : v0/v1 = rows (k0+2*half)/(k0+2*half+1), N = lane%16
//   C 16x16: VGPR r -> row r (lanes 0-15) / row r+8 (lanes 16-31)
// ---------------------------------------------------------------------------
template <int NT, bool RELU>
__global__ __launch_bounds__(256) void k_gemm(const float* __restrict__ A,
                                              const float* __restrict__ Wp,
                                              const float* __restrict__ bias,
                                              float* __restrict__ C, int N) {
  __shared__ float As[MBLK * LDA];
  const int tid   = threadIdx.x;
  const int mbase = blockIdx.x * MBLK;

  stage_A(A, As, mbase, tid);

  const int wave = tid >> 5;   // 0..7 -> M tile
  const int lane = tid & 31;
  const int half = lane >> 4;  // selects K pair
  const int nn   = lane & 15;

  v8f acc[NT];
#pragma unroll
  for (int j = 0; j < NT; ++j)
    acc[j] = (v8f){0.f, 0.f, 0.f, 0.f, 0.f, 0.f, 0.f, 0.f};

  const float* Arow = &As[(wave * 16 + nn) * LDA];

  for (int k0 = 0; k0 < EMB; k0 += 4) {
    const int kk = k0 + 2 * half;
    const v2f a = *(const v2f*)(Arow + kk);
#pragma unroll
    for (int j = 0; j < NT; ++j) {
      const v2f b = *(const v2f*)(Wp + (j * 16 + nn) * EMB + kk);
      acc[j] = wmma4(a, b, acc[j]);
    }
  }

#pragma unroll
  for (int j = 0; j < NT; ++j) {
    const int col = j * 16 + nn;
    if (col < N) {
      const float bv = bias ? bias[col] : 0.f;
#pragma unroll
      for (int r = 0; r < 8; ++r) {
        const int row = mbase + wave * 16 + r + 8 * half;
        float v = acc[j][r] + bv;
        if (RELU) v = fmaxf(v, 0.f);
        C[(size_t)row * N + col] = v;
      }
    }
  }
}

// ---------------------------------------------------------------------------
// Attention for one (b,h): scores = scale * K Q^T, causal mask (keep s<=t),
// softmax over the *t* axis (faithful to the reference's axis=2 softmax),
// attn = P V. One 64-thread block per (b,h); k,v and P staged in LDS.
// ---------------------------------------------------------------------------
__global__ __launch_bounds__(64) void k_attn(const float* __restrict__ kb,
                                             const float* __restrict__ qb,
                                             const float* __restrict__ vb,
                                             float* __restrict__ out) {
  const int b = blockIdx.x, h = blockIdx.y;
  __shared__ float ks[TT * HS];
  __shared__ float vs[TT * HS];
  __shared__ float ps[TT * 52];   // P[t][s], stride 52
  const int tid = threadIdx.x;

  for (int i = tid; i < TT * HS; i += 64) {
    int t = i / HS, d = i - t * HS;
    size_t g = (size_t)(b * TT + t) * EMB + h * HS + d;
#if USE_ASYNC_LDS
    __builtin_amdgcn_global_load_async_to_lds_b32(
        (ASG int*)(kb + g), (ASL int*)&ks[i], 0, 0);
    __builtin_amdgcn_global_load_async_to_lds_b32(
        (ASG int*)(vb + g), (ASL int*)&vs[i], 0, 0);
#else
    ks[i] = kb[g];
    vs[i] = vb[g];
#endif
  }
#if USE_ASYNC_LDS
  __builtin_amdgcn_s_wait_asynccnt(0);
#endif
  __syncthreads();

  if (tid < TT) {
    const int s = tid;
    float q0, q1, q2, q3, q4;
    {
      size_t g = (size_t)(b * TT + s) * EMB + h * HS;
      q0 = qb[g]; q1 = qb[g + 1]; q2 = qb[g + 2]; q3 = qb[g + 3]; q4 = qb[g + 4];
    }
    const float scale = 10.0f;  // sqrt(E) = sqrt(100)
    // pass 1: column max over valid t in [s, T)
    float m = -INFINITY;
    for (int t = s; t < TT; ++t) {
      const float* kr = &ks[t * HS];
      float w = scale * (kr[0]*q0 + kr[1]*q1 + kr[2]*q2 + kr[3]*q3 + kr[4]*q4);
      m = fmaxf(m, w);
    }
    // pass 2: exp + column sum, write unnormalized into LDS (0 for t<s)
    float den = 0.f;
    for (int t = 0; t < TT; ++t) {
      float e = 0.f;
      if (t >= s) {
        const float* kr = &ks[t * HS];
        float w = scale * (kr[0]*q0 + kr[1]*q1 + kr[2]*q2 + kr[3]*q3 + kr[4]*q4);
        e = __expf(w - m);
        den += e;
      }
      ps[t * 52 + s] = e;
    }
    const float inv = 1.0f / den;
    for (int t = s; t < TT; ++t) ps[t * 52 + s] *= inv;
  }
  __syncthreads();

  if (tid < TT) {
    const int t = tid;
    float a0 = 0.f, a1 = 0.f, a2 = 0.f, a3 = 0.f, a4 = 0.f;
    const float* pr = &ps[t * 52];
    for (int s = 0; s < TT; ++s) {
      const float p = pr[s];
      const float* vr = &vs[s * HS];
      a0 += p * vr[0]; a1 += p * vr[1]; a2 += p * vr[2];
      a3 += p * vr[3]; a4 += p * vr[4];
    }
    size_t g = (size_t)(b * TT + t) * EMB + h * HS;
    out[g] = a0; out[g + 1] = a1; out[g + 2] = a2; out[g + 3] = a3; out[g + 4] = a4;
  }
}

// ---------------------------------------------------------------------------
// LM head + fused log-softmax/NLL. Same WMMA GEMM (NT=5, Wp padded to 80
// rows); logits go to d_out and to LDS (aliasing the dead A-stage buffer),
// then per-row logsumexp and a deterministic per-block tree reduction.
// ---------------------------------------------------------------------------
__global__ __launch_bounds__(256) void k_lm_loss(const float* __restrict__ A,
                                                 const float* __restrict__ Wp,
                                                 const float* __restrict__ bias,
                                                 const int* __restrict__ targets,
                                                 float* __restrict__ logits,
                                                 float* __restrict__ partial) {
  __shared__ float smem[MBLK * LDA];  // A stage, reused as logits[128][80]
  __shared__ float red[MBLK];
  const int tid   = threadIdx.x;
  const int mbase = blockIdx.x * MBLK;

  stage_A(A, smem, mbase, tid);

  const int wave = tid >> 5;
  const int lane = tid & 31;
  const int half = lane >> 4;
  const int nn   = lane & 15;
  constexpr int NT = 5;

  v8f acc[NT];
#pragma unroll
  for (int j = 0; j < NT; ++j)
    acc[j] = (v8f){0.f, 0.f, 0.f, 0.f, 0.f, 0.f, 0.f, 0.f};

  const float* Arow = &smem[(wave * 16 + nn) * LDA];
  for (int k0 = 0; k0 < EMB; k0 += 4) {
    const int kk = k0 + 2 * half;
    const v2f a = *(const v2f*)(Arow + kk);
#pragma unroll
    for (int j = 0; j < NT; ++j) {
      const v2f b = *(const v2f*)(Wp + (j * 16 + nn) * EMB + kk);
      acc[j] = wmma4(a, b, acc[j]);
    }
  }
  __syncthreads();  // all waves done reading the A stage; alias smem as logits

#pragma unroll
  for (int j = 0; j < NT; ++j) {
    const int col = j * 16 + nn;
    if (col < VOC) {
      const float bv = bias[col];
#pragma unroll
      for (int r = 0; r < 8; ++r) {
        const int rin = wave * 16 + r + 8 * half;
        const float v = acc[j][r] + bv;
        logits[(size_t)(mbase + rin) * VOC + col] = v;
        smem[rin * 80 + col] = v;
      }
    }
  }
  __syncthreads();

  if (tid < MBLK) {
    const float* rowp = &smem[tid * 80];
    float m = -INFINITY;
    for (int j = 0; j < VOC; ++j) m = fmaxf(m, rowp[j]);
    float s = 0.f;
    for (int j = 0; j < VOC; ++j) s += __expf(rowp[j] - m);
    const float lse = m + __logf(s);
    const int tg = targets[mbase + tid];
    red[tid] = lse - rowp[tg];
  }
  __syncthreads();
  for (int off = 64; off > 0; off >>= 1) {
    if (tid < off) red[tid] += red[tid + off];
    __syncthreads();
  }
  if (tid == 0) partial[blockIdx.x] = red[0];
}

__global__ void k_loss(const float* __restrict__ partial, float* __restrict__ out) {
  if (blockIdx.x == 0 && threadIdx.x == 0) {
    float s = 0.f;
    for (int i = 0; i < NBLOCKS; ++i) s += partial[i];  // fixed order: deterministic
    out[(size_t)BT * VOC] = s / (float)BT;
  }
}

// ---------------------------------------------------------------------------
extern "C" void kernel_launch(void* const* d_in, const int* in_sizes, int n_in,
                              void* d_out, int out_size, void* d_ws, size_t ws_size,
                              hipStream_t stream) {
  const int*   idx     = (const int*)d_in[0];
  const int*   targets = (const int*)d_in[1];
  const float* tok     = (const float*)d_in[2];
  const float* pos     = (const float*)d_in[3];
  const float* Wk      = (const float*)d_in[4];   // [H,D,E] -> flat [100,100]
  const float* Wq      = (const float*)d_in[5];
  const float* Wv      = (const float*)d_in[6];
  const float* Wff     = (const float*)d_in[7];   // [E,E]
  const float* bff     = (const float*)d_in[8];
  const float* Wlm     = (const float*)d_in[9];   // [V,E]
  const float* blm     = (const float*)d_in[10];
  float* out = (float*)d_out;
  float* ws  = (float*)d_ws;

  const size_t S = (size_t)BT * EMB;  // 10.24M floats per activation buffer
  float* x       = ws;                // [BT,E]
  float* kb      = ws + S;
  float* qb      = ws + 2 * S;
  float* vb      = ws + 3 * S;
  float* attn    = x;                 // x dead after qkv projections
  float* hbuf    = kb;                // k dead after attention
  float* partial = ws + 4 * S;        // NBLOCKS floats
  // zero-padded weights (rows padded to NT*16 so GEMM inner loop is branch-free)
  float* wpk  = partial + NBLOCKS;    // 112 x 100
  float* wpq  = wpk + 112 * EMB;
  float* wpv  = wpq + 112 * EMB;
  float* wpff = wpv + 112 * EMB;
  float* wplm = wpff + 112 * EMB;     // 80 x 100

  const int padElems7 = 112 * EMB, padElems5 = 80 * EMB;
  k_padw<<<(padElems7 + 255) / 256, 256, 0, stream>>>(Wk,  wpk,  EMB, 112);
  k_padw<<<(padElems7 + 255) / 256, 256, 0, stream>>>(Wq,  wpq,  EMB, 112);
  k_padw<<<(padElems7 + 255) / 256, 256, 0, stream>>>(Wv,  wpv,  EMB, 112);
  k_padw<<<(padElems7 + 255) / 256, 256, 0, stream>>>(Wff, wpff, EMB, 112);
  k_padw<<<(padElems5 + 255) / 256, 256, 0, stream>>>(Wlm, wplm, VOC, 80);

  k_embed<<<(BT * EMB + 255) / 256, 256, 0, stream>>>(idx, tok, pos, x);

  k_gemm<7, false><<<NBLOCKS, 256, 0, stream>>>(x, wpk, nullptr, kb, EMB);
  k_gemm<7, false><<<NBLOCKS, 256, 0, stream>>>(x, wpq, nullptr, qb, EMB);
  k_gemm<7, false><<<NBLOCKS, 256, 0, stream>>>(x, wpv, nullptr, vb, EMB);

  k_attn<<<dim3(BB, HEADS), 64, 0, stream>>>(kb, qb, vb, attn);

  k_gemm<7, true><<<NBLOCKS, 256, 0, stream>>>(attn, wpff, bff, hbuf, EMB);

  k_lm_loss<<<NBLOCKS, 256, 0, stream>>>(hbuf, wplm, blm, targets, out, partial);
  k_loss<<<1, 1, 0, stream>>>(partial, out);
}